// QLSTMCell_39779987095929
// MI455X (gfx1250) — compile-verified
//
#include <hip/hip_runtime.h>
#include <hip/hip_bf16.h>
#include <math.h>

typedef __attribute__((ext_vector_type(2))) float v2f;
typedef __attribute__((ext_vector_type(8))) float v8f;

#define BATCH   16384
#define DIN     128
#define NQ      10
#define NSTATES 1024   // 2^10
#define BSTRIDE 132    // padded LDS row stride (dwords) -> conflict-free b64 reads

// ---------------------------------------------------------------------------
// Kernel A: h = x @ W^T + b  via V_WMMA_F32_16X16X4_F32 (f32 exact path).
// 8 waves / block, each wave owns one 16-row M tile; N padded 10 -> 16 via a
// zero-padded B tile staged in LDS (removes per-iteration EXEC divergence).
// All 32 B fragments are preloaded to VGPRs; inner loop = A load + WMMA only.
// A layout (ISA 7.12.2, 32-bit A 16x4): lane%16 = M, K = vgpr + 2*(lane/16).
// B layout (mirror, 4x16):              lane%16 = N, K = vgpr + 2*(lane/16).
// C/D (32-bit 16x16): lane%16 = N, M = vgpr + 8*(lane/16).
// ---------------------------------------------------------------------------
__global__ __launch_bounds__(256) void qlstm_gemm(const float* __restrict__ x,
                                                  const float* __restrict__ W,
                                                  const float* __restrict__ bias,
                                                  float* __restrict__ h) {
    __shared__ float wlds[16 * BSTRIDE];   // zero-padded W (16 x 128), ~8.4 KB

    // cooperative zero-padded stage of W into LDS
    for (int i = threadIdx.x; i < 16 * DIN; i += 256) {
        const int n = i / DIN, k = i % DIN;
        wlds[n * BSTRIDE + k] = (n < NQ) ? W[n * DIN + k] : 0.f;
    }
    __syncthreads();

    const int lane    = threadIdx.x & 31;
    const int wave    = threadIdx.x >> 5;
    const int mbase   = (blockIdx.x * 8 + wave) * 16;
    const int halfSel = lane >> 4;       // 0 = lanes 0-15, 1 = lanes 16-31
    const int lid     = lane & 15;
    const int arow    = mbase + lid;     // A-matrix row held by this lane
    const int n       = lid;             // B/C column held by this lane

    const float* __restrict__ xrow = x + (size_t)arow * DIN;
    const float* __restrict__ brow = &wlds[n * BSTRIDE + 2 * halfSel];

    // preload all B fragments (unconditional ds_load_b64, uniform EXEC)
    v2f breg[DIN / 4];
#pragma unroll
    for (int kk = 0; kk < DIN / 4; ++kk)
        breg[kk] = *(const v2f*)(brow + kk * 4);

    v8f c = {};
#pragma unroll
    for (int kk = 0; kk < DIN / 4; ++kk) {
        const int k0 = kk * 4 + 2 * halfSel;          // even -> 8B aligned
        if (kk + 2 < DIN / 4)
            __builtin_prefetch(xrow + (kk + 2) * 4, 0, 0);   // global_prefetch_b8
        v2f a = *(const v2f*)(xrow + k0);
        // (neg_a, A, neg_b, B, c_mod, C, reuse_a, reuse_b)
        c = __builtin_amdgcn_wmma_f32_16x16x4_f32(false, a, false, breg[kk],
                                                  (short)0, c, false, false);
    }

    if (n < NQ) {
        const float bn = bias[n];
#pragma unroll
        for (int v = 0; v < 8; ++v) {
            const int m = mbase + v + 8 * halfSel;
            h[(size_t)m * NQ + n] = c[v] + bn;
        }
    }
}

// ---------------------------------------------------------------------------
// Kernel B: four 10-qubit VQCs per batch row, state resident in LDS (8 KB),
// fused LSTM gating tail. Wire w acts on bit (9 - w) (wire 0 = MSB).
// ---------------------------------------------------------------------------
__global__ __launch_bounds__(256) void qlstm_vqc(const float* __restrict__ h,
                                                 const float* __restrict__ wf,
                                                 const float* __restrict__ wi,
                                                 const float* __restrict__ wu,
                                                 const float* __restrict__ wo,
                                                 float* __restrict__ out) {
    __shared__ float re[NSTATES];
    __shared__ float im[NSTATES];
    __shared__ float hsh[NQ];
    __shared__ float exp4[4][NQ];

    const int tid = threadIdx.x;
    const int b   = blockIdx.x;

    if (tid < NQ)      hsh[tid]            = h[(size_t)b * NQ + tid];
    if (tid < 4 * NQ)  ((float*)exp4)[tid] = 0.f;
    __syncthreads();

    for (int g = 0; g < 4; ++g) {
        // ingate<-wf(RX), forgetgate<-wi(RY), cellgate<-wu(RX), outgate<-wo(RX)
        const float* __restrict__ wptr =
            (g == 0) ? wf : (g == 1) ? wi : (g == 2) ? wu : wo;
        const bool isRY = (g == 1);

        // init |0...0>
        for (int k = tid; k < NSTATES; k += 256) {
            re[k] = (k == 0) ? 1.f : 0.f;
            im[k] = 0.f;
        }
        __syncthreads();

        // 20 single-qubit gates: 10 RX (embedding, per-row angle) then
        // 10 RX/RY (shared layer weights)
        for (int gg = 0; gg < 2 * NQ; ++gg) {
            const int   q     = (gg < NQ) ? gg : (gg - NQ);
            const bool  doRY  = (gg >= NQ) && isRY;
            const float theta = (gg < NQ) ? hsh[q] : wptr[q];
            float sn, cs;
            __sincosf(0.5f * theta, &sn, &cs);

            const int p    = 9 - q;
            const int mask = 1 << p;
            for (int t = tid; t < NSTATES / 2; t += 256) {
                const int k0 = ((t & ~(mask - 1)) << 1) | (t & (mask - 1));
                const int k1 = k0 | mask;
                const float a0r = re[k0], a0i = im[k0];
                const float a1r = re[k1], a1i = im[k1];
                if (doRY) {     // [[c,-s],[s,c]] real
                    re[k0] = cs * a0r - sn * a1r;  im[k0] = cs * a0i - sn * a1i;
                    re[k1] = sn * a0r + cs * a1r;  im[k1] = sn * a0i + cs * a1i;
                } else {        // RX: [[c,-is],[-is,c]]
                    re[k0] = cs * a0r + sn * a1i;  im[k0] = cs * a0i - sn * a1r;
                    re[k1] = cs * a1r + sn * a0i;  im[k1] = cs * a1i - sn * a0r;
                }
            }
            __syncthreads();
        }

        // ring of CNOTs: control wire i, target wire (i+1)%10
        for (int i = 0; i < NQ; ++i) {
            const int cb  = 9 - i;
            const int tb  = 9 - ((i + 1) % NQ);
            const int lo  = cb < tb ? cb : tb;
            const int hi  = cb < tb ? tb : cb;
            const int mlo = (1 << lo) - 1;
            const int mhi = (1 << hi) - 1;
            // 256 swap pairs: insert zeros at lo,hi; set control=1; flip target
            const int t  = tid;
            const int u  = ((t & ~mlo) << 1) | (t & mlo);
            const int k  = ((u & ~mhi) << 1) | (u & mhi);
            const int ka = k | (1 << cb);        // control=1, target=0
            const int kb = ka | (1 << tb);       // control=1, target=1
            const float tr = re[ka], ti = im[ka];
            re[ka] = re[kb];  im[ka] = im[kb];
            re[kb] = tr;      im[kb] = ti;
            __syncthreads();
        }

        // PauliZ expvals: sign(w,k) = +1 if bit (9-w) of k is 0 else -1
        float acc[NQ];
#pragma unroll
        for (int w = 0; w < NQ; ++w) acc[w] = 0.f;
        for (int k = tid; k < NSTATES; k += 256) {
            const float pr = re[k] * re[k] + im[k] * im[k];
#pragma unroll
            for (int w = 0; w < NQ; ++w)
                acc[w] += ((k >> (9 - w)) & 1) ? -pr : pr;
        }
#pragma unroll
        for (int w = 0; w < NQ; ++w) {
            float v = acc[w];
            for (int off = 16; off > 0; off >>= 1)
                v += __shfl_down(v, off, 32);
            if ((tid & 31) == 0) atomicAdd(&exp4[g][w], v);   // ds_add_f32
        }
        __syncthreads();
    }

    // fused LSTM cell tail
    if (tid < NQ) {
        const float ing  = 1.f / (1.f + __expf(-exp4[0][tid]));
        const float forg = 1.f / (1.f + __expf(-exp4[1][tid]));
        const float cell = tanhf(exp4[2][tid]);
        const float outg = 1.f / (1.f + __expf(-exp4[3][tid]));
        const float hn   = hsh[tid] * forg + ing * cell;
        out[(size_t)b * NQ + tid] = outg * tanhf(hn);
    }
}

// ---------------------------------------------------------------------------
extern "C" void kernel_launch(void* const* d_in, const int* in_sizes, int n_in,
                              void* d_out, int out_size, void* d_ws, size_t ws_size,
                              hipStream_t stream) {
    const float* x    = (const float*)d_in[0];   // (16384,128)
    const float* W    = (const float*)d_in[1];   // (10,128)
    const float* bias = (const float*)d_in[2];   // (10,)
    const float* wf   = (const float*)d_in[3];   // (1,10)
    const float* wi   = (const float*)d_in[4];
    const float* wu   = (const float*)d_in[5];
    const float* wo   = (const float*)d_in[6];
    float*       out  = (float*)d_out;           // (16384,10)
    float*       hbuf = (float*)d_ws;            // 16384*10 f32 = 640 KB scratch

    (void)in_sizes; (void)n_in; (void)out_size; (void)ws_size;

    qlstm_gemm<<<BATCH / (16 * 8), 256, 0, stream>>>(x, W, bias, hbuf);
    qlstm_vqc <<<BATCH,            256, 0, stream>>>(hbuf, wf, wi, wu, wo, out);
}